// Cross_Attention_8143257993388
// MI455X (gfx1250) — compile-verified
//
#include <hip/hip_runtime.h>
#include <hip/hip_bf16.h>

typedef float v2f __attribute__((ext_vector_type(2)));
typedef float v8f __attribute__((ext_vector_type(8)));

#define DIMC 384
#define HEADS 8
#define CH 48
#define QDIM 324
#define SQ 18
#define IH 128
#define IW 128
#define NPIX (IH*IW)      // 16384
#define NB 8
#define GRD 72            // gathered k-grid per axis
#define NG (GRD*GRD)      // 5184
#define KC 32             // GEMM k-chunk

// --------------------------------------------------------------------------
// Async global->LDS helpers (CDNA5 GLOBAL_LOAD_ASYNC_TO_LDS_*, ASYNCcnt).
// LDS operand: low 32 bits of the generic pointer == wave-relative LDS addr.
// --------------------------------------------------------------------------
__device__ __forceinline__ void async_ld_b32(uint32_t lds, const float* g) {
    asm volatile("global_load_async_to_lds_b32 %0, %1, off"
                 :: "v"(lds), "v"(g) : "memory");
}
__device__ __forceinline__ void async_ld_b128(uint32_t lds, const float* g) {
    asm volatile("global_load_async_to_lds_b128 %0, %1, off"
                 :: "v"(lds), "v"(g) : "memory");
}
__device__ __forceinline__ uint32_t lds_off(const void* p) {
    return (uint32_t)(uintptr_t)p;
}

// ---------------------------------------------------------------------------
// Setup: bilinear indices/fractions, gather column indices, folded proj PW.
// ---------------------------------------------------------------------------
__global__ void setup_kernel(const float* __restrict__ proj_w,
                             float* __restrict__ PW,     // [384*8]
                             int*   __restrict__ y0arr,  // [18]
                             float* __restrict__ farr,   // [18]
                             int*   __restrict__ colidx) // [5184]
{
    int t = threadIdx.x; // 256 threads, one block
    if (t < SQ) {
        float src = (t + 0.5f) * (128.0f / 18.0f) - 0.5f;
        if (src < 0.f) src = 0.f;
        int i0 = (int)floorf(src);
        if (i0 > IH - 1) i0 = IH - 1;
        y0arr[t] = i0;
        farr[t]  = src - (float)i0;
    }
    __syncthreads();
    for (int g = t; g < NG; g += blockDim.x) {
        int ri = g / GRD, ci = g % GRD;
        int iy = y0arr[ri >> 2] - 1 + (ri & 3);
        int ix = y0arr[ci >> 2] - 1 + (ci & 3);
        iy = min(max(iy, 0), IH - 1);
        ix = min(max(ix, 0), IW - 1);
        colidx[g] = iy * IW + ix;
    }
    for (int oh = t; oh < DIMC * HEADS; oh += blockDim.x) {
        int o = oh >> 3, hh = oh & 7;
        const float* pr = proj_w + o * DIMC + hh * CH;
        float s = 0.f;
        for (int c = 0; c < CH; ++c) s += pr[c];
        PW[oh] = s;
    }
}

// ---------------------------------------------------------------------------
// qn[b,i] = normalize_rows(query @ Wq^T)
// ---------------------------------------------------------------------------
__global__ void qn_kernel(const float* __restrict__ query,
                          const float* __restrict__ Wq,
                          float* __restrict__ qn)
{
    __shared__ float red[512];
    int b = blockIdx.x;
    int t = threadIdx.x; // 352 threads
    float q = 0.f;
    if (t < QDIM) {
        const float* wr = Wq + (size_t)t * QDIM;
        const float* qr = query + (size_t)b * QDIM;
        for (int j = 0; j < QDIM; ++j) q += qr[j] * wr[j];
    }
    red[t] = (t < QDIM) ? q * q : 0.f;
    if (t < 512 - 352) red[352 + t] = 0.f;
    __syncthreads();
    for (int s = 256; s > 0; s >>= 1) { if (t < s) red[t] += red[t + s]; __syncthreads(); }
    float nrm = fmaxf(sqrtf(red[0]), 1e-12f);
    if (t < QDIM) qn[(size_t)b * QDIM + t] = q / nrm;
}

// ---------------------------------------------------------------------------
// fp32 WMMA GEMM with async-LDS double-buffered staging of X.
//   C[m,n] = sum_k Wm[m*K+k] * X[k*ldx + col(n)],  col(n)=colidx?colidx[n]:n
// Block = 256 thr = 8 waves; wave w -> rows m0=bx*128+w*16, cols n0=by*64..+63.
// X chunk (32k x 64n) staged TRANSPOSED in LDS: Xs[n*34 + k] so each B
// fragment is one aligned 8B LDS read, bank-conflict free.
// ---------------------------------------------------------------------------
__global__ __launch_bounds__(256)
void gemm16_wmma(const float* __restrict__ Wm,   // M x K, row-major
                 const float* __restrict__ X,    // rows stride ldx (per batch)
                 float* __restrict__ C,          // M x N (per batch), ldc = N
                 const int* __restrict__ colidx,
                 int K, int N, int ldx,
                 long long xBatchStride, long long cBatchStride)
{
    __shared__ float Xs[2][64 * 34];
    int tid  = threadIdx.x;
    int lane = tid & 31;
    int wv   = tid >> 5;
    int b    = blockIdx.z;
    const float* Xb = X + (long long)b * xBatchStride;
    float*       Cb = C + (long long)b * cBatchStride;

    int m0 = blockIdx.x * 128 + wv * 16;
    int n0 = blockIdx.y * 64;
    int j  = lane & 15;
    int kh = (lane >> 4) * 2;
    const float* Arow = Wm + (size_t)(m0 + j) * K + kh;

    // staging map: element (k = p*4 + tid>>6, n = tid&63) for p = 0..7
    int sn = tid & 63;
    int sk = tid >> 6;                                   // 0..3
    int scol = colidx ? colidx[n0 + sn] : (n0 + sn);
    const float* gbase = Xb + scol;
    uint32_t l0 = lds_off(&Xs[0][sn * 34 + sk]);
    uint32_t l1 = lds_off(&Xs[1][sn * 34 + sk]);

    auto stage = [&](int k0, int buf) {
        uint32_t lb = buf ? l1 : l0;
#pragma unroll
        for (int p = 0; p < 8; ++p) {
            const float* g = gbase + (long long)(k0 + p * 4 + sk) * ldx;
            async_ld_b32(lb + (uint32_t)(p * 4 * sizeof(float)), g);
        }
    };

    stage(0, 0);
    v8f acc[4] = {};
    for (int k0 = 0; k0 < K; k0 += KC) {
        int buf = (k0 / KC) & 1;
        bool more = (k0 + KC) < K;
        if (more) {
            stage(k0 + KC, buf ^ 1);
            asm volatile("s_wait_asynccnt 0x8" ::: "memory"); // chunk k0 landed
        } else {
            asm volatile("s_wait_asynccnt 0x0" ::: "memory");
        }
        __syncthreads();
        const float* ar = Arow + k0;
        const float* bs = &Xs[buf][0];
#pragma unroll
        for (int s = 0; s < 8; ++s) {
            v2f a;
            a.x = ar[4 * s];
            a.y = ar[4 * s + 1];
#pragma unroll
            for (int t = 0; t < 4; ++t) {
                const float* bp = bs + (t * 16 + j) * 34 + 4 * s + kh;
                v2f bb;
                bb.x = bp[0];
                bb.y = bp[1];
                acc[t] = __builtin_amdgcn_wmma_f32_16x16x4_f32(
                    false, a, false, bb, (short)0, acc[t], false, false);
            }
        }
        __syncthreads();   // all waves done reading buf before it is restaged
    }

    int mo = m0 + (lane >> 4) * 8;
#pragma unroll
    for (int r = 0; r < 8; ++r)
#pragma unroll
        for (int t = 0; t < 4; ++t)
            Cb[(long long)(mo + r) * N + n0 + t * 16 + j] = acc[t][r];
}

// ---------------------------------------------------------------------------
// k path: 3x3 depthwise on the gathered 72x72 grid, bilinear 18x18,
// then L2-normalize the 324-vector -> kn[b,c,324].
// ---------------------------------------------------------------------------
__global__ void kres_kernel(const float* __restrict__ kvk,   // [B,384,5184]
                            const float* __restrict__ dw_w,  // k channels: c*9
                            const float* __restrict__ farr,
                            float* __restrict__ kn)
{
    __shared__ float g[NG];
    __shared__ float red[512];
    int c = blockIdx.x;
    int b = blockIdx.y;
    int t = threadIdx.x; // 352
    const float* src = kvk + ((long long)b * DIMC + c) * NG;
    for (int i = t; i < NG; i += 352) g[i] = src[i];
    float w9[9];
    const float* wp = dw_w + (size_t)c * 9;
#pragma unroll
    for (int q = 0; q < 9; ++q) w9[q] = wp[q];
    __syncthreads();

    float kr = 0.f;
    if (t < SQ * SQ) {
        int i = t / SQ, jj = t % SQ;
        float fy = farr[i], fx = farr[jj];
        float d2[2][2];
#pragma unroll
        for (int a = 0; a < 2; ++a) {
#pragma unroll
            for (int e = 0; e < 2; ++e) {
                int gr = 4 * i + 1 + a, gc = 4 * jj + 1 + e;
                float d = 0.f;
#pragma unroll
                for (int dy = -1; dy <= 1; ++dy)
#pragma unroll
                    for (int dx = -1; dx <= 1; ++dx)
                        d += w9[(dy + 1) * 3 + (dx + 1)] * g[(gr + dy) * GRD + (gc + dx)];
                d2[a][e] = d;
            }
        }
        kr = (1.f - fy) * ((1.f - fx) * d2[0][0] + fx * d2[0][1]) +
             fy         * ((1.f - fx) * d2[1][0] + fx * d2[1][1]);
    }
    red[t] = (t < SQ * SQ) ? kr * kr : 0.f;
    if (t < 512 - 352) red[352 + t] = 0.f;
    __syncthreads();
    for (int s = 256; s > 0; s >>= 1) { if (t < s) red[t] += red[t + s]; __syncthreads(); }
    float nrm = fmaxf(sqrtf(red[0]), 1e-12f);
    if (t < SQ * SQ) kn[((long long)b * DIMC + c) * (SQ * SQ) + t] = kr / nrm;
}

// ---------------------------------------------------------------------------
// A[b,h,d] = softmax_d( temp[h] * dot(qn[b,:], kn[b,h*48+d,:]) )
// ---------------------------------------------------------------------------
__global__ void attn_kernel(const float* __restrict__ qn,
                            const float* __restrict__ kn,
                            const float* __restrict__ temperature,
                            float* __restrict__ A)
{
    __shared__ float sc[CH];
    int hh = blockIdx.x, b = blockIdx.y;
    int t = threadIdx.x; // 256
    int lane = t & 31, wv = t >> 5;
    const float* qb = qn + (size_t)b * QDIM;
    float tmp = temperature[hh];
    for (int d = wv; d < CH; d += 8) {
        const float* kr = kn + (((long long)b * HEADS + hh) * CH + d) * QDIM;
        float s = 0.f;
        for (int l = lane; l < QDIM; l += 32) s += qb[l] * kr[l];
#pragma unroll
        for (int off = 16; off > 0; off >>= 1) s += __shfl_xor(s, off, 32);
        if (lane == 0) sc[d] = s * tmp;
    }
    __syncthreads();
    if (t == 0) {
        float mx = sc[0];
        for (int d = 1; d < CH; ++d) mx = fmaxf(mx, sc[d]);
        float sum = 0.f, e[CH];
        for (int d = 0; d < CH; ++d) { e[d] = expf(sc[d] - mx); sum += e[d]; }
        float inv = 1.f / sum;
        float* Ab = A + ((long long)b * HEADS + hh) * CH;
        for (int d = 0; d < CH; ++d) Ab[d] = e[d] * inv;
    }
}

// ---------------------------------------------------------------------------
// Fused depthwise-3x3 + O accumulation + folded projection, reading vkv
// directly (no materialized v).  Tile: 8 rows x 64 cols per block; channels
// staged 16 at a time (10 halo rows x 72 halo cols) via async b128 copies.
//   O[h,pix]   = sum_d A[b,h,d] * dw3x3(vkv[b, h*48+d])[pix]
//   out[b,o,s] = sum_h PW[o,h] * O[h,s]
// ---------------------------------------------------------------------------
#define CCH 16
#define LROW 72
__global__ __launch_bounds__(256)
void fused_out_kernel(const float* __restrict__ vkv,
                      const float* __restrict__ dw_w,
                      const float* __restrict__ A,
                      const float* __restrict__ PW,
                      float* __restrict__ out)
{
    __shared__ float Ls[CCH * 10 * LROW];     // 46 KB
    __shared__ float coef[DIMC];
    __shared__ float pws[DIMC * HEADS];
    __shared__ float dws[DIMC * 9];

    int tid = threadIdx.x;           // 256
    int b   = blockIdx.y;
    int yt  = blockIdx.x & 15;       // 16 row tiles
    int xt  = blockIdx.x >> 4;       // 2 col tiles

    const float* Ab = A + (long long)b * DIMC;
    for (int i = tid; i < DIMC; i += 256) coef[i] = Ab[i];
    for (int i = tid; i < DIMC * HEADS; i += 256) pws[i] = PW[i];
    for (int i = tid; i < DIMC * 9; i += 256) dws[i] = dw_w[(size_t)DIMC * 9 + i]; // v-channel weights
    __syncthreads();

    int lc = tid & 63;               // local col 0..63
    int lr0 = tid >> 6;              // local rows lr0 and lr0+4

    float O[HEADS][2];
#pragma unroll
    for (int h = 0; h < HEADS; ++h) { O[h][0] = 0.f; O[h][1] = 0.f; }

    const float* vb = vkv + (long long)b * DIMC * NPIX;

#pragma unroll
    for (int h = 0; h < HEADS; ++h) {
        for (int s = 0; s < 3; ++s) {
            int cg0 = h * CH + s * CCH;
            // ---- stage CCH x 10 x 72 halo tile (async b128, zero OOB) ----
            for (int p = 0; p < 12; ++p) {
                int f = p * 256 + tid;           // quad id over 160 rows x 18 quads
                if (f < CCH * 10 * 18) {
                    int row_id = f / 18;
                    int q18    = f - row_id * 18;
                    int c  = row_id / 10;
                    int rr = row_id - c * 10;
                    int gy  = yt * 8 - 1 + rr;
                    int gx0 = xt * 64 - 4 + q18 * 4;
                    float* dst = Ls + (row_id * LROW + q18 * 4);
                    if (gy >= 0 && gy < IH && gx0 >= 0 && gx0 <= IW - 4) {
                        const float* g = vb + ((long long)(cg0 + c) * NPIX + gy * IW + gx0);
                        async_ld_b128(lds_off(dst), g);
                    } else {
                        dst[0] = 0.f; dst[1] = 0.f; dst[2] = 0.f; dst[3] = 0.f;
                    }
                }
            }
            asm volatile("s_wait_asynccnt 0x0" ::: "memory");
            __syncthreads();
            // ---- accumulate O over this channel chunk ----
            for (int c = 0; c < CCH; ++c) {
                int cg = cg0 + c;
                float cf = coef[cg];
                const float* w9 = dws + cg * 9;
                const float* lb = Ls + c * 10 * LROW;
#pragma unroll
                for (int q = 0; q < 2; ++q) {
                    int lr = 4 * q + lr0;
                    const float* r0 = lb + (lr + 0) * LROW + lc + 4; // dy=-1
                    const float* r1 = lb + (lr + 1) * LROW + lc + 4; // dy= 0
                    const float* r2 = lb + (lr + 2) * LROW + lc + 4; // dy=+1
                    float d = w9[0] * r0[-1] + w9[1] * r0[0] + w9[2] * r0[1]
                            + w9[3] * r1[-1] + w9[4] * r1[0] + w9[5] * r1[1]
                            + w9[6] * r2[-1] + w9[7] * r2[0] + w9[8] * r2[1];
                    O[h][q] += cf * d;
                }
            }
            __syncthreads();   // before restaging Ls
        }
    }

    // ---- folded projection: out[b,o,pix] = sum_h PW[o,h] * O[h,pix] ----
    int gy0 = yt * 8 + lr0;
    int gx  = xt * 64 + lc;
    float* ob = out + (long long)b * DIMC * NPIX;
    for (int o = 0; o < DIMC; ++o) {
        const float* pw = pws + o * HEADS;
        float r0 = 0.f, r1 = 0.f;
#pragma unroll
        for (int h = 0; h < HEADS; ++h) {
            float w = pw[h];
            r0 += w * O[h][0];
            r1 += w * O[h][1];
        }
        float* orow = ob + (long long)o * NPIX;
        orow[gy0 * IW + gx]       = r0;
        orow[(gy0 + 4) * IW + gx] = r1;
    }
}

// ---------------------------------------------------------------------------
extern "C" void kernel_launch(void* const* d_in, const int* in_sizes, int n_in,
                              void* d_out, int out_size, void* d_ws, size_t ws_size,
                              hipStream_t stream) {
    (void)in_sizes; (void)n_in; (void)out_size; (void)ws_size;
    const float* x      = (const float*)d_in[0];
    const float* query  = (const float*)d_in[1];
    const float* Wq     = (const float*)d_in[2];
    const float* kv_w   = (const float*)d_in[3];
    const float* dw_w   = (const float*)d_in[4];
    const float* proj_w = (const float*)d_in[5];
    const float* temp   = (const float*)d_in[6];
    float* out = (float*)d_out;

    float* ws = (float*)d_ws;
    size_t o = 0;
    float* vkv = ws + o;  o += (size_t)NB * DIMC * NPIX;   // 50.3M floats
    float* kvk = ws + o;  o += (size_t)NB * DIMC * NG;     // 15.9M floats
    float* qn  = ws + o;  o += (size_t)NB * QDIM;
    float* kn  = ws + o;  o += (size_t)NB * DIMC * SQ * SQ;
    float* Aw  = ws + o;  o += (size_t)NB * HEADS * CH;
    float* PW  = ws + o;  o += (size_t)DIMC * HEADS;
    float* farr = ws + o; o += SQ;
    int* y0arr = (int*)(ws + o);  o += SQ;
    int* colidx = (int*)(ws + o); o += NG;

    setup_kernel<<<1, 256, 0, stream>>>(proj_w, PW, y0arr, farr, colidx);
    qn_kernel<<<NB, 352, 0, stream>>>(query, Wq, qn);

    // v-path 1x1 conv: rows 384..767 of kv_w
    dim3 g2(DIMC / 128, NPIX / 64, NB);
    gemm16_wmma<<<g2, 256, 0, stream>>>(kv_w + (size_t)DIMC * DIMC, x, vkv, nullptr,
                                        DIMC, NPIX, NPIX,
                                        (long long)DIMC * NPIX, (long long)DIMC * NPIX);
    // k-path 1x1 conv at gathered 72x72 grid: rows 0..383 of kv_w
    dim3 g3(DIMC / 128, NG / 64, NB);
    gemm16_wmma<<<g3, 256, 0, stream>>>(kv_w, x, kvk, colidx,
                                        DIMC, NG, NPIX,
                                        (long long)DIMC * NPIX, (long long)DIMC * NG);

    dim3 g4(DIMC, NB);
    kres_kernel<<<g4, 352, 0, stream>>>(kvk, dw_w, farr, kn);

    dim3 g5(HEADS, NB);
    attn_kernel<<<g5, 256, 0, stream>>>(qn, kn, temp, Aw);

    dim3 g7(32, NB);   // 2 col tiles x 16 row tiles
    fused_out_kernel<<<g7, 256, 0, stream>>>(vkv, dw_w, Aw, PW, out);
}